// LengthRegulator_80728205295984
// MI455X (gfx1250) — compile-verified
//
#include <hip/hip_runtime.h>
#include <stdint.h>

#define B_ 32
#define T_ 1024
#define C_ 384
#define MAXLEN_ 7168            // T * (DUR_MAX-1)
#define FRAMES_PER_BLOCK 128    // 8 waves x 16 frames

typedef float v4f __attribute__((ext_vector_type(4)));

// ---------------------------------------------------------------------------
// Kernel 1: per-batch inclusive scan of sanitized durations (Hillis-Steele in
// LDS), emit cum[] to workspace and mel_lens (= max(cum[-1],1)) both as int
// (workspace, for kernel 2) and as float (tail of d_out, tuple output #2).
// ---------------------------------------------------------------------------
__global__ __launch_bounds__(1024) void lr_cumsum_kernel(
    const int* __restrict__ dur, int* __restrict__ cum,
    int* __restrict__ mel, float* __restrict__ mel_f)
{
    __shared__ int s[T_];
    const int t = threadIdx.x;
    const int b = blockIdx.x;

    int d = dur[b * T_ + t];
    d = (d >= 0 && d <= 1000) ? d : 0;   // sanitize exactly like reference
    s[t] = d;
    __syncthreads();

#pragma unroll
    for (int off = 1; off < T_; off <<= 1) {
        int v = (t >= off) ? s[t - off] : 0;
        __syncthreads();
        s[t] += v;
        __syncthreads();
    }

    cum[b * T_ + t] = s[t];
    if (t == T_ - 1) {
        int m = (s[t] < 1) ? 1 : s[t];
        mel[b]   = m;
        mel_f[b] = (float)m;
    }
}

// ---------------------------------------------------------------------------
// Kernel 2: expansion. Stage cum[b][0..T) into LDS via CDNA5 async
// global->LDS B128 copies, then one frame per wave32: uniform binary search
// in LDS + 3x B128 gather per lane (C=384 = 32 lanes * 3 * float4), written
// with non-temporal stores (352MB write-once stream > 192MB L2; keep the
// re-read 48MB x table L2-resident instead).
// ---------------------------------------------------------------------------
__global__ __launch_bounds__(256) void lr_expand_kernel(
    const float* __restrict__ x, const int* __restrict__ cum,
    const int* __restrict__ mel, float* __restrict__ out)
{
    __shared__ int scum[T_];
    const int tid = threadIdx.x;           // 0..255
    const int b   = blockIdx.y;

    // --- CDNA5 async tensor-family copy: 256 lanes x 16B = 4KB of cum[b] ---
    {
        const int* cbase = cum + b * T_;                  // uniform -> SGPR pair
        unsigned voff = (unsigned)(tid * 16);             // per-lane byte offset
        unsigned ldsa = (unsigned)(uintptr_t)scum + voff; // LDS byte address
        asm volatile(
            "global_load_async_to_lds_b128 %0, %1, %2\n\t"
            "s_wait_asynccnt 0"
            :: "v"(ldsa), "v"(voff), "s"(cbase)
            : "memory");
    }
    __syncthreads();   // every wave waited its own ASYNCcnt; barrier publishes

    const int mel_b = mel[b];
    const int wave  = tid >> 5;
    const int lane  = tid & 31;

    const v4f* __restrict__ xb = (const v4f*)(x   + (size_t)b * T_ * C_);
    v4f*       __restrict__ ob = (v4f*)      (out + (size_t)b * MAXLEN_ * C_);

    const int j_base = blockIdx.x * FRAMES_PER_BLOCK;

#pragma unroll 2
    for (int i = 0; i < FRAMES_PER_BLOCK / 8; ++i) {
        const int j = j_base + wave + (i << 3);   // wave-uniform frame index

        v4f* dst = ob + (size_t)j * (C_ / 4);

        if (j < mel_b) {                          // uniform branch per wave
            // searchsorted(cum, j, side='right') over 1024 elems: 10 steps
            int lo = 0, hi = T_;
#pragma unroll
            for (int it = 0; it < 10; ++it) {
                const int  mid = (lo + hi) >> 1;
                const bool c   = (scum[mid] <= j);   // broadcast LDS read
                lo = c ? (mid + 1) : lo;
                hi = c ? hi : mid;
            }
            const int idx = (lo < (T_ - 1)) ? lo : (T_ - 1);

            const v4f* src = xb + (size_t)idx * (C_ / 4);
#pragma unroll
            for (int k = 0; k < 3; ++k) {
                v4f v = src[lane + 32 * k];
                __builtin_nontemporal_store(v, &dst[lane + 32 * k]);
            }
        } else {
            v4f z = (v4f)0.0f;
#pragma unroll
            for (int k = 0; k < 3; ++k)
                __builtin_nontemporal_store(z, &dst[lane + 32 * k]);
        }
    }
}

// ---------------------------------------------------------------------------
// d_in[0] = x          float32 [B,T,C]
// d_in[1] = durations  int32   [B,T]
// d_out   = stacked float32 [B,MAXLEN,C] ++ mel_lens [B] (flat, return order)
// d_ws    = cum int32 [B,T] ++ mel int32 [B]
// ---------------------------------------------------------------------------
extern "C" void kernel_launch(void* const* d_in, const int* in_sizes, int n_in,
                              void* d_out, int out_size, void* d_ws, size_t ws_size,
                              hipStream_t stream)
{
    (void)in_sizes; (void)n_in; (void)out_size; (void)ws_size;

    const float* x   = (const float*)d_in[0];
    const int*   dur = (const int*)d_in[1];
    float*       out = (float*)d_out;

    int* cum = (int*)d_ws;
    int* mel = cum + (size_t)B_ * T_;
    float* mel_f = out + (size_t)B_ * MAXLEN_ * C_;

    lr_cumsum_kernel<<<dim3(B_), dim3(T_), 0, stream>>>(dur, cum, mel, mel_f);

    lr_expand_kernel<<<dim3(MAXLEN_ / FRAMES_PER_BLOCK, B_), dim3(256), 0, stream>>>(
        x, cum, mel, out);
}